// Router_20306605375573
// MI455X (gfx1250) — compile-verified
//
#include <hip/hip_runtime.h>
#include <hip/hip_bf16.h>
#include <math.h>

// ---------------------------------------------------------------------------
// MoE router: logits = h @ W^T  (T=16384, D=4096, E=64), softmax, gumbel,
// top-K=8 mask.  Memory-bound (~290MB @ 23.3TB/s => ~12.5us), so we keep fp32
// end-to-end and use the native CDNA5 fp32 WMMA (V_WMMA_F32_16X16X4_F32).
// Outputs (flat, in order): mask (T*E), probs (T*E), logits_clean (T*E),
// logits_sel (T*E) -- all float32.
// ---------------------------------------------------------------------------

typedef __attribute__((ext_vector_type(2))) float v2f;
typedef __attribute__((ext_vector_type(8))) float v8f;

#define RT_T 16384
#define RT_D 4096
#define RT_E 64
#define RT_K 8

// ---------------------------------------------------------------------------
// Kernel 1: skinny GEMM via fp32 WMMA.
// One wave per 16-row tile of tokens; each wave produces a 16x64 logits tile
// as four 16x16 WMMA accumulators, looping K=4096 in steps of 4.
//
// A-operand layout (16x4 f32, 2 VGPRs): lane l holds M=l%16;
//   VGPR0 = K = 2*(l/16), VGPR1 = K = 2*(l/16)+1.
// B-operand layout (4x16 f32, 2 VGPRs): lane l holds N=l%16;
//   VGPR0 = K = 2*(l/16), VGPR1 = K = 2*(l/16)+1.
// C/D layout (16x16 f32, 8 VGPRs): element (M = v + 8*(l/16), N = l%16).
// ---------------------------------------------------------------------------
__global__ __launch_bounds__(256) void router_gemm_wmma(
    const float* __restrict__ h,   // (T, D)
    const float* __restrict__ W,   // (E, D)
    float* __restrict__ logits)    // (T, E)
{
  const int gwave = (int)((blockIdx.x * blockDim.x + threadIdx.x) >> 5);
  const int lane  = (int)(threadIdx.x & 31);
  const int row0  = gwave * 16;
  if (row0 >= RT_T) return;

  const int m    = lane & 15;        // row / col within 16-tile
  const int half = lane >> 4;        // 0 or 1
  const int kh   = half * 2;         // K offset within group-of-4

  const float* __restrict__ ha = h + (size_t)(row0 + m) * RT_D + kh;
  const float* __restrict__ w0 = W + (size_t)( 0 + m) * RT_D + kh;
  const float* __restrict__ w1 = W + (size_t)(16 + m) * RT_D + kh;
  const float* __restrict__ w2 = W + (size_t)(32 + m) * RT_D + kh;
  const float* __restrict__ w3 = W + (size_t)(48 + m) * RT_D + kh;

  v8f acc0 = {}, acc1 = {}, acc2 = {}, acc3 = {};

#pragma unroll 4
  for (int k = 0; k < RT_D; k += 4) {
    const v2f a  = *(const v2f*)(ha + k);
    const v2f b0 = *(const v2f*)(w0 + k);
    const v2f b1 = *(const v2f*)(w1 + k);
    const v2f b2 = *(const v2f*)(w2 + k);
    const v2f b3 = *(const v2f*)(w3 + k);
    acc0 = __builtin_amdgcn_wmma_f32_16x16x4_f32(false, a, false, b0,
                                                 (short)0, acc0, false, false);
    acc1 = __builtin_amdgcn_wmma_f32_16x16x4_f32(false, a, false, b1,
                                                 (short)0, acc1, false, false);
    acc2 = __builtin_amdgcn_wmma_f32_16x16x4_f32(false, a, false, b2,
                                                 (short)0, acc2, false, false);
    acc3 = __builtin_amdgcn_wmma_f32_16x16x4_f32(false, a, false, b3,
                                                 (short)0, acc3, false, false);
  }

  // Write back per C/D layout: M = v + 8*half, N = (lane&15) + 16*tile.
  const int n = lane & 15;
#pragma unroll
  for (int v = 0; v < 8; ++v) {
    const size_t r = (size_t)(row0 + v + 8 * half) * RT_E;
    logits[r + n +  0] = acc0[v];
    logits[r + n + 16] = acc1[v];
    logits[r + n + 32] = acc2[v];
    logits[r + n + 48] = acc3[v];
  }
}

// ---------------------------------------------------------------------------
// Kernel 2: per-row softmax, gumbel-noised select logits, top-8 mask.
// One wave32 per token row; each lane owns experts {lane, lane+32}.
// Tie-break in top-k: lowest index wins (matches jax.lax.top_k).
// ---------------------------------------------------------------------------
__global__ __launch_bounds__(256) void router_post(
    const float* __restrict__ logits,  // (T, E) clean
    const float* __restrict__ u,       // (T, E) uniform noise
    float* __restrict__ mask,          // (T, E)
    float* __restrict__ probs,         // (T, E)
    float* __restrict__ lsel)          // (T, E)
{
  const int lane = (int)(threadIdx.x & 31);
  const int row  = (int)(blockIdx.x * (blockDim.x >> 5) + (threadIdx.x >> 5));
  if (row >= RT_T) return;

  const size_t base = (size_t)row * RT_E;
  const float l0 = logits[base + lane];
  const float l1 = logits[base + lane + 32];

  // ---- softmax over 64 experts (router_temp = 1) ----
  float mx = fmaxf(l0, l1);
#pragma unroll
  for (int off = 16; off > 0; off >>= 1)
    mx = fmaxf(mx, __shfl_xor(mx, off, 32));
  const float e0 = __expf(l0 - mx);
  const float e1 = __expf(l1 - mx);
  float s = e0 + e1;
#pragma unroll
  for (int off = 16; off > 0; off >>= 1)
    s += __shfl_xor(s, off, 32);
  const float inv = 1.0f / s;
  probs[base + lane]      = e0 * inv;
  probs[base + lane + 32] = e1 * inv;

  // ---- gumbel-noised selection logits (tau = 1, select_temp = 1) ----
  const float g0 = -__logf(-__logf(u[base + lane]));
  const float g1 = -__logf(-__logf(u[base + lane + 32]));
  float s0 = l0 + g0;
  float s1 = l1 + g1;
  lsel[base + lane]      = s0;
  lsel[base + lane + 32] = s1;

  // ---- top-8 via iterative argmax with masking ----
  bool sel0 = false, sel1 = false;
#pragma unroll
  for (int t = 0; t < RT_K; ++t) {
    const float v0 = sel0 ? -INFINITY : s0;
    const float v1 = sel1 ? -INFINITY : s1;
    float v;
    int   idx;
    if (v0 >= v1) { v = v0; idx = lane; }         // prefer lower index on tie
    else          { v = v1; idx = lane + 32; }
#pragma unroll
    for (int off = 16; off > 0; off >>= 1) {
      const float ov = __shfl_xor(v,   off, 32);
      const int   oi = __shfl_xor(idx, off, 32);
      if (ov > v || (ov == v && oi < idx)) { v = ov; idx = oi; }
    }
    if      (idx == lane)      sel0 = true;
    else if (idx == lane + 32) sel1 = true;
  }
  mask[base + lane]      = sel0 ? 1.0f : 0.0f;
  mask[base + lane + 32] = sel1 ? 1.0f : 0.0f;
}

// ---------------------------------------------------------------------------
extern "C" void kernel_launch(void* const* d_in, const int* in_sizes, int n_in,
                              void* d_out, int out_size, void* d_ws, size_t ws_size,
                              hipStream_t stream) {
  const float* h = (const float*)d_in[0];   // (T, D)
  const float* W = (const float*)d_in[1];   // (E, D)
  const float* u = (const float*)d_in[2];   // (T, E)

  float* out    = (float*)d_out;
  const size_t TE = (size_t)RT_T * RT_E;
  float* mask   = out;
  float* probs  = out + TE;
  float* logits = out + 2 * TE;   // logits_clean slice of the output tuple
  float* lsel   = out + 3 * TE;

  // GEMM: one wave per 16 rows -> T/16 = 1024 waves; 8 waves (256 thr) / block.
  {
    dim3 block(256);
    dim3 grid((RT_T / 16 + 7) / 8);
    router_gemm_wmma<<<grid, block, 0, stream>>>(h, W, logits);
  }
  // Post: one wave per row -> 8 rows per 256-thread block.
  {
    dim3 block(256);
    dim3 grid(RT_T / 8);
    router_post<<<grid, block, 0, stream>>>(logits, u, mask, probs, lsel);
  }
}